// MeanAggregator_27358941675829
// MI455X (gfx1250) — compile-verified
//
#include <hip/hip_runtime.h>
#include <hip/hip_bf16.h>

typedef __attribute__((ext_vector_type(2))) float v2f;
typedef __attribute__((ext_vector_type(8))) float v8f;

#define D 64
#define LDS_PITCH 65   // avoid 64-bank conflicts on column reads

// ---------------------------------------------------------------------------
// Zero a contiguous float region (count divisible by 4).
// ---------------------------------------------------------------------------
__global__ void MeanAgg_zero_kernel(float4* __restrict__ p, long n4) {
    long i = (long)blockIdx.x * blockDim.x + threadIdx.x;
    long stride = (long)gridDim.x * blockDim.x;
    float4 z = make_float4(0.f, 0.f, 0.f, 0.f);
    for (; i < n4; i += stride) p[i] = z;
}

// ---------------------------------------------------------------------------
// new_emb[m][n] = sum_k feature_b[m][k] * W[n][k] + bias[n]
// One wave computes a 16-row x 64-col strip via 4x (16x16) WMMA accumulators,
// K=64 swept in 16 steps of V_WMMA_F32_16X16X4_F32.
// Block = 256 threads = 8 waves = 128 rows per block.
// ---------------------------------------------------------------------------
__global__ void MeanAgg_gemm_kernel(const float* __restrict__ fb,
                                    const float* __restrict__ W,
                                    const float* __restrict__ bias,
                                    float* __restrict__ emb,
                                    int nb) {
    __shared__ float ldsW[D * LDS_PITCH];

    // cooperative load of W (64x64) into LDS with padded pitch
    for (int t = threadIdx.x; t < D * D; t += blockDim.x) {
        int r = t >> 6, c = t & 63;
        ldsW[r * LDS_PITCH + c] = W[t];
    }
    __syncthreads();

    const int lane = threadIdx.x & 31;
    const int wave = threadIdx.x >> 5;
    const int row0 = (blockIdx.x * 8 + wave) * 16;
    if (row0 >= nb) return;                 // uniform per-wave exit

    const int half = lane >> 4;             // 0: K={0,1}, 1: K={2,3} within a K4 group
    const int l16  = lane & 15;

    // Clamp tail rows so EXEC stays all-ones through WMMA; store is guarded.
    int arow = row0 + l16;
    if (arow >= nb) arow = nb - 1;
    const float* arow_p = fb + (long)arow * D;

    // A fragments for the whole K=64 sweep (16 x v2f = 32 VGPRs)
    v2f afrag[16];
#pragma unroll
    for (int k0 = 0; k0 < 16; ++k0) {
        const float* ap = arow_p + k0 * 4 + half * 2;
        afrag[k0].x = ap[0];
        afrag[k0].y = ap[1];
    }

    v8f acc[4] = {v8f{}, v8f{}, v8f{}, v8f{}};

#pragma unroll
    for (int j = 0; j < 4; ++j) {           // N tiles: columns j*16 .. j*16+15
        const int n = j * 16 + l16;         // this lane's column within tile
        v2f bfrag[16];
#pragma unroll
        for (int k0 = 0; k0 < 16; ++k0) {   // B[k][n] = W[n][k]
            const float* bp = &ldsW[n * LDS_PITCH + k0 * 4 + half * 2];
            bfrag[k0].x = bp[0];
            bfrag[k0].y = bp[1];
        }
#pragma unroll
        for (int k0 = 0; k0 < 16; ++k0) {
            acc[j] = __builtin_amdgcn_wmma_f32_16x16x4_f32(
                false, afrag[k0], false, bfrag[k0],
                (short)0, acc[j], false, false);
        }
    }

    // store: acc[j] VGPR i holds element (row0 + i + half*8, j*16 + l16)
#pragma unroll
    for (int j = 0; j < 4; ++j) {
        const int n = j * 16 + l16;
        const float bb = bias[n];
#pragma unroll
        for (int i = 0; i < 8; ++i) {
            int m = row0 + i + half * 8;
            if (m < nb) emb[(long)m * D + n] = acc[j][i] + bb;
        }
    }
}

// ---------------------------------------------------------------------------
// Per-edge gather + scatter-add. One wave per edge (grid-stride over edges):
// lane reads dims {lane, lane+32} of emb[dst], atomic-adds into summed[src].
// Lane 0 bumps the degree counter.
// ---------------------------------------------------------------------------
__global__ void MeanAgg_scatter_kernel(const int* __restrict__ src,
                                       const int* __restrict__ dst,
                                       const float* __restrict__ emb,
                                       float* __restrict__ summed,
                                       float* __restrict__ deg,
                                       int E) {
    const int lane = threadIdx.x & 31;
    int wid = (int)((blockIdx.x * blockDim.x + threadIdx.x) >> 5);
    const int nwaves = (int)((gridDim.x * blockDim.x) >> 5);
    for (int e = wid; e < E; e += nwaves) {
        const int s = src[e];
        const int d = dst[e];
        const float* row = emb + (long)d * D;
        float v0 = row[lane];
        float v1 = row[lane + 32];
        float* orow = summed + (long)s * D;
        atomicAdd(&orow[lane], v0);
        atomicAdd(&orow[lane + 32], v1);
        if (lane == 0) atomicAdd(&deg[s], 1.0f);
    }
}

// ---------------------------------------------------------------------------
// out = summed / max(deg, 1-if-zero). float4 per thread (16 per row).
// ---------------------------------------------------------------------------
__global__ void MeanAgg_finalize_kernel(const float4* __restrict__ summed,
                                        const float* __restrict__ deg,
                                        float4* __restrict__ out,
                                        long n4) {
    long i = (long)blockIdx.x * blockDim.x + threadIdx.x;
    if (i >= n4) return;
    float dg = deg[i >> 4];                 // 16 float4 per 64-wide row
    if (dg == 0.0f) dg = 1.0f;
    float inv = 1.0f / dg;
    float4 v = summed[i];
    v.x *= inv; v.y *= inv; v.z *= inv; v.w *= inv;
    out[i] = v;
}

// ---------------------------------------------------------------------------
extern "C" void kernel_launch(void* const* d_in, const int* in_sizes, int n_in,
                              void* d_out, int out_size, void* d_ws, size_t ws_size,
                              hipStream_t stream) {
    const int*   src  = (const int*)d_in[0];
    const int*   dst  = (const int*)d_in[1];
    const float* fb   = (const float*)d_in[2];
    const float* W    = (const float*)d_in[3];
    const float* bias = (const float*)d_in[4];

    const int E  = in_sizes[0];
    const int NB = in_sizes[2] / D;
    const int NA = out_size / D;

    float* ws     = (float*)d_ws;
    float* emb    = ws;                          // NB * 64
    float* summed = ws + (size_t)NB * D;         // NA * 64
    float* deg    = summed + (size_t)NA * D;     // NA

    // 1) zero summed + deg (contiguous region)
    {
        long n4 = ((long)NA * D + (long)NA) / 4;
        int blocks = (int)((n4 + 255) / 256);
        if (blocks > 2048) blocks = 2048;
        MeanAgg_zero_kernel<<<blocks, 256, 0, stream>>>((float4*)summed, n4);
    }

    // 2) GEMM: emb = fb @ W^T + bias  (WMMA fp32)
    {
        int blocks = (NB + 127) / 128;           // 128 rows per block (8 waves x 16)
        MeanAgg_gemm_kernel<<<blocks, 256, 0, stream>>>(fb, W, bias, emb, NB);
    }

    // 3) gather + scatter-add (atomic f32, L2-resident target)
    {
        MeanAgg_scatter_kernel<<<2048, 256, 0, stream>>>(src, dst, emb, summed, deg, E);
    }

    // 4) divide by clamped degree into d_out
    {
        long n4 = (long)NA * D / 4;
        int blocks = (int)((n4 + 255) / 256);
        MeanAgg_finalize_kernel<<<blocks, 256, 0, stream>>>(
            (const float4*)summed, deg, (float4*)d_out, n4);
    }
}